// VIN_68401649156499
// MI455X (gfx1250) — compile-verified
//
#include <hip/hip_runtime.h>

typedef float v2f __attribute__((ext_vector_type(2)));
typedef float v4f __attribute__((ext_vector_type(4)));
typedef float v8f __attribute__((ext_vector_type(8)));

#define HW    64
#define STR   66                  // halo-padded row stride
#define PLANE (STR * STR)         // 4356 floats per halo plane
#define NPIX  4096
#define NB    128
#define QSIZE (NB * 5 * NPIX)

// ---------------------------------------------------------------------------
// Prep kernel: fuse h->r convs (linear) and build the padded B-matrix table.
// ws layout (floats):
//   [0..17]  w_eff[ch][tap]  = sum_c w_r[c] * w_h[c,ch,tap]
//   [18]     b_eff           = sum_c w_r[c] * b_h[c]
//   [32 + kk*16 + n]  Wtab, kk=0..19 (taps 0..8 = w_q row, 9..17 = w_w row,
//                      18..19 = zero pad), n=0..15 (channels, >=5 zero)
// ---------------------------------------------------------------------------
__global__ __launch_bounds__(64) void vin_prep(const float* __restrict__ w_h,
                                               const float* __restrict__ b_h,
                                               const float* __restrict__ w_r,
                                               const float* __restrict__ w_q,
                                               const float* __restrict__ w_w,
                                               float* __restrict__ ws) {
  int t = threadIdx.x;
  if (t < 18) {
    int ch = t / 9, k = t % 9;
    float acc = 0.f;
    for (int c = 0; c < 150; ++c) acc += w_r[c] * w_h[c * 18 + ch * 9 + k];
    ws[t] = acc;
  } else if (t == 18) {
    float acc = 0.f;
    for (int c = 0; c < 150; ++c) acc += w_r[c] * b_h[c];
    ws[18] = acc;
  }
  for (int idx = t; idx < 20 * 16; idx += 64) {
    int kk = idx >> 4, n = idx & 15;
    float v = 0.f;
    if (n < 5) {
      if (kk < 9)       v = w_q[n * 9 + kk];
      else if (kk < 18) v = w_w[n * 9 + (kk - 9)];
    }
    ws[32 + idx] = v;
  }
}

// im2col A-row -> LDS offset within rv_s (plane 0 = r, plane 1 = v).
// kk>=18 is clamped to 17: the matching B rows are zero, so the loaded
// value is multiplied by 0 and irrelevant -> unconditional load, no branch.
__device__ __forceinline__ int vin_aoff(int kk) {
  kk = (kk > 17) ? 17 : kk;
  int pl = (kk < 9) ? 0 : 1;
  int ts = (kk < 9) ? kk : kk - 9;
  return pl * PLANE + (ts / 3) * STR + (ts % 3);
}

// One 16-pixel conv group: 5 chained V_WMMA_F32_16X16X4_F32 (K = 18 + pad2).
__device__ __forceinline__ v8f vin_conv_group(const float* rv_s, int base,
                                              const int* aoff0, const int* aoff1,
                                              const v2f* breg) {
  v8f c = {};
#pragma unroll
  for (int k = 0; k < 5; ++k) {
    v2f a;
    a.x = rv_s[base + aoff0[k]];
    a.y = rv_s[base + aoff1[k]];
    c = __builtin_amdgcn_wmma_f32_16x16x4_f32(
            false, a, false, breg[k], (short)0, c, false, false);
  }
  return c;
}

// ---------------------------------------------------------------------------
// Main kernel: one workgroup per batch image; entire 81-step value-iteration
// recurrence runs out of LDS.
// ---------------------------------------------------------------------------
__global__ __launch_bounds__(256) void vin_iterate(const float* __restrict__ inp,
                                                   const int* __restrict__ coords,
                                                   const float* __restrict__ ws,
                                                   float* __restrict__ out) {
  __shared__ float rv_s[2 * PLANE];    // [0,PLANE)=r plane, [PLANE,2*PLANE)=v
  __shared__ float q_s[5 * NPIX];      // per-channel q planes (80 KB)

  const int b    = blockIdx.x;
  const int t    = threadIdx.x;
  const int lane = t & 31;
  const int wave = t >> 5;
  const int half = (lane >> 4) & 1;    // K-pair selector within the wave
  const int n    = lane & 15;          // A-row pixel index / D-column channel

  // Zero both halo planes (borders stay zero => SAME padding for free).
  for (int i = t; i < 2 * PLANE; i += 256) rv_s[i] = 0.f;
  __syncthreads();

  // Fused conv weights (uniform).
  float weff[18];
#pragma unroll
  for (int i = 0; i < 18; ++i) weff[i] = ws[i];
  const float beff = ws[18];

  // Loop-invariant B operands: chunk k covers im2col rows 4k..4k+3;
  // this lane holds rows (4k+2*half, 4k+2*half+1), column n.
  v2f breg[5];
#pragma unroll
  for (int k = 0; k < 5; ++k) {
    int r0 = 4 * k + 2 * half;
    breg[k].x = ws[32 + r0 * 16 + n];
    breg[k].y = ws[32 + (r0 + 1) * 16 + n];
  }

  // Loop-invariant per-lane A-operand LDS offsets.
  int aoff0[5], aoff1[5];
#pragma unroll
  for (int k = 0; k < 5; ++k) {
    aoff0[k] = vin_aoff(4 * k + 2 * half);
    aoff1[k] = vin_aoff(4 * k + 2 * half + 1);
  }

  const int sx  = coords[b * 4 + 0];
  const int sy  = coords[b * 4 + 1];
  const int tgt = sx * 64 + sy;        // logits gather pixel

  // r = conv3x3(input, w_eff) + b_eff  (the fused h->r stage).
  const float* in0 = inp + (size_t)b * 2 * NPIX;
  for (int i = 0; i < 16; ++i) {
    int p = t + 256 * i;
    int y0 = p >> 6, x0 = p & 63;
    float acc = beff;
#pragma unroll
    for (int ch = 0; ch < 2; ++ch) {
#pragma unroll
      for (int tap = 0; tap < 9; ++tap) {
        int yy = y0 + tap / 3 - 1, xx = x0 + tap % 3 - 1;
        if (yy >= 0 && yy < 64 && xx >= 0 && xx < 64)
          acc += weff[ch * 9 + tap] * in0[ch * NPIX + yy * 64 + xx];
      }
    }
    rv_s[(y0 + 1) * STR + (x0 + 1)] = acc;
  }
  __syncthreads();

  // ---- 80 recurrence steps, entirely in LDS (no global traffic) ----------
  for (int it = 0; it < 80; ++it) {
    for (int gi = 0; gi < 32; ++gi) {
      int p0   = (wave * 32 + gi) * 16;
      int base = (p0 >> 6) * STR + (p0 & 63) + n;
      v8f c = vin_conv_group(rv_s, base, aoff0, aoff1, breg);
      // D layout: VGPR j / this lane = Q[pixel p0+j+8*half, channel n]
      if (n < 5) {
        float* dst = &q_s[n * NPIX + p0 + 8 * half];
        *(v4f*)dst       = (v4f){c[0], c[1], c[2], c[3]};
        *(v4f*)(dst + 4) = (v4f){c[4], c[5], c[6], c[7]};
      }
    }
    __syncthreads();
    // v = max over the 5 channels, written into the halo-padded v plane.
    for (int i = 0; i < 16; ++i) {
      int p = t + 256 * i;
      float v = q_s[p];
#pragma unroll
      for (int ch = 1; ch < 5; ++ch) v = fmaxf(v, q_s[ch * NPIX + p]);
      rv_s[PLANE + (p >> 6) * STR + (p & 63) + STR + 1] = v;
    }
    __syncthreads();
  }

  // ---- Final q = conv([r, v_79], ker): store to global + gather logits ---
  float* out_q = out + (size_t)b * 5 * NPIX;
  float* out_l = out + (size_t)QSIZE + b * 5;
  for (int gi = 0; gi < 32; ++gi) {
    int p0   = (wave * 32 + gi) * 16;
    int base = (p0 >> 6) * STR + (p0 & 63) + n;
    v8f c = vin_conv_group(rv_s, base, aoff0, aoff1, breg);
    if (n < 5) {
      float* dst = out_q + n * NPIX + p0 + 8 * half;
      *(v4f*)dst       = (v4f){c[0], c[1], c[2], c[3]};
      *(v4f*)(dst + 4) = (v4f){c[4], c[5], c[6], c[7]};
      int rel = tgt - (p0 + 8 * half);
      if (rel >= 0 && rel < 8) out_l[n] = c[rel];
    }
  }
}

extern "C" void kernel_launch(void* const* d_in, const int* in_sizes, int n_in,
                              void* d_out, int out_size, void* d_ws, size_t ws_size,
                              hipStream_t stream) {
  const float* input_view = (const float*)d_in[0];
  const int*   coords     = (const int*)  d_in[1];
  const float* w_h        = (const float*)d_in[2];
  const float* b_h        = (const float*)d_in[3];
  const float* w_r        = (const float*)d_in[4];
  const float* w_q        = (const float*)d_in[5];
  const float* w_w        = (const float*)d_in[6];
  float* out = (float*)d_out;
  float* ws  = (float*)d_ws;

  vin_prep<<<1, 64, 0, stream>>>(w_h, b_h, w_r, w_q, w_w, ws);
  vin_iterate<<<NB, 256, 0, stream>>>(input_view, coords, ws, out);
}